// VariableSelectionNetwork_3959959847520
// MI455X (gfx1250) — compile-verified
//
#include <hip/hip_runtime.h>
#include <hip/hip_bf16.h>
#include <stdint.h>

// Problem constants (match reference)
#define BT 8
#define ST 512
#define FD 128
#define HD 128
#define NT (BT*ST)      // 4096 tokens
#define LNEPS 1e-5f

// Expert kernel tiling
#define WAVES 8
#define TOK_PER_WG (WAVES*16)   // 128 tokens per workgroup
#define FPG 16                  // features per workgroup (feature group)
#define WSTRIDE 132             // padded LDS row stride (floats): 128+4 -> bank-conflict-free

typedef __attribute__((ext_vector_type(2)))  float v2f;
typedef __attribute__((ext_vector_type(8)))  float v8f;
typedef __attribute__((ext_vector_type(4)))  unsigned int v4u;
typedef __attribute__((ext_vector_type(8)))  int v8i;
typedef __attribute__((ext_vector_type(4)))  int v4i;

__device__ __forceinline__ float elu1(float v) { return v > 0.f ? v : (__expf(v) - 1.f); }
__device__ __forceinline__ float sigm(float v) { return 1.f / (1.f + __expf(-v)); }

// ---------------------------------------------------------------------------
// Kernel A: zero the 'selected' output region (it is accumulated with atomics)
// ---------------------------------------------------------------------------
__global__ void vsn_zero_kernel(float* __restrict__ p, int n) {
    int i = blockIdx.x * blockDim.x + threadIdx.x;
    if (i < n) p[i] = 0.f;
}

// ---------------------------------------------------------------------------
// Kernel B: selection GRN -> softmax weights [NT, FD]  (1 block of 128 thr / token)
// ---------------------------------------------------------------------------
__global__ __launch_bounds__(128)
void vsn_sel_kernel(const float* __restrict__ x,
                    const float* __restrict__ w1, const float* __restrict__ b1,
                    const float* __restrict__ w2, const float* __restrict__ b2,
                    const float* __restrict__ wg, const float* __restrict__ bg,
                    const float* __restrict__ lng, const float* __restrict__ lnb,
                    float* __restrict__ wout) {
    __shared__ float xb[FD], hb[HD], h2b[HD], red[8];
    const int t = blockIdx.x;
    const int j = threadIdx.x;
    const int wv = j >> 5, ln = j & 31;

    xb[j] = x[(size_t)t * FD + j];
    __syncthreads();

    float acc = b1[j];
    for (int i = 0; i < FD; ++i) acc += xb[i] * w1[j * FD + i];
    hb[j] = elu1(acc);
    __syncthreads();

    acc = b2[j];
    for (int i = 0; i < HD; ++i) acc += hb[i] * w2[j * HD + i];
    h2b[j] = acc;
    __syncthreads();

    float ga = bg[j];
    for (int i = 0; i < HD; ++i) ga += h2b[i] * wg[j * HD + i];
    const float pre = sigm(ga) * h2b[j] + xb[j];

    // LayerNorm over 128
    float s = pre, s2 = pre * pre;
    for (int m = 16; m >= 1; m >>= 1) { s += __shfl_xor(s, m, 32); s2 += __shfl_xor(s2, m, 32); }
    if (ln == 0) { red[wv] = s; red[4 + wv] = s2; }
    __syncthreads();
    s  = red[0] + red[1] + red[2] + red[3];
    s2 = red[4] + red[5] + red[6] + red[7];
    const float mean = s * (1.f / FD);
    const float var  = s2 * (1.f / FD) - mean * mean;
    const float lno  = (pre - mean) * __frsqrt_rn(var + LNEPS) * lng[j] + lnb[j];
    __syncthreads();

    // softmax over 128
    float mx = lno;
    for (int m = 16; m >= 1; m >>= 1) mx = fmaxf(mx, __shfl_xor(mx, m, 32));
    if (ln == 0) red[wv] = mx;
    __syncthreads();
    mx = fmaxf(fmaxf(red[0], red[1]), fmaxf(red[2], red[3]));
    const float e = __expf(lno - mx);
    float se = e;
    for (int m = 16; m >= 1; m >>= 1) se += __shfl_xor(se, m, 32);
    __syncthreads();
    if (ln == 0) red[4 + wv] = se;
    __syncthreads();
    se = red[4] + red[5] + red[6] + red[7];
    wout[(size_t)t * FD + j] = e / se;
}

// ---------------------------------------------------------------------------
// TDM helper: DMA one 128x128 f32 matrix from global to LDS with +4 DWORD row
// padding (LDS row stride becomes 132 floats). Issued once per wave.
// ---------------------------------------------------------------------------
__device__ __forceinline__ void tdm_load_128x128(const float* gsrc, unsigned lds_byte_addr) {
    const uint64_t ga = (uint64_t)(uintptr_t)gsrc;
    v4u g0;
    g0[0] = 1u;                       // count=1, user descriptor
    g0[1] = lds_byte_addr;            // lds_addr
    g0[2] = (unsigned)ga;             // global_addr[31:0]
    g0[3] = (unsigned)((ga >> 32) & 0x1FFFFFFull) | 0x80000000u; // addr[56:32] | type=2
    v8i g1;
    // workgroup_mask=0 | data_size=2(4B) | pad_enable | pad_interval=6(128 dw) | pad_amount=3(4 dw)
    g1[0] = (int)((2u << 16) | (1u << 20) | (6u << 22) | (3u << 25));
    g1[1] = (int)(HD << 16);          // tensor_dim0 = 128 (low16)
    g1[2] = (int)(HD << 16);          // tensor_dim0 hi16=0 | tensor_dim1=128 (low16)
    g1[3] = (int)(HD << 16);          // tensor_dim1 hi16=0 | tile_dim0 = 128
    g1[4] = (int)HD;                  // tile_dim1 = 128, tile_dim2 = 0
    g1[5] = (int)HD;                  // tensor_dim0_stride = 128 (low32)
    g1[6] = (int)((HD * HD) << 16);   // stride hi16=0 | tensor_dim1_stride low16 = 16384
    g1[7] = 0;                        // tensor_dim1_stride hi32 = 0
    v4i g2 = {0, 0, 0, 0};            // dims 3/4 unused (2D tile)
    v4i g3 = {0, 0, 0, 0};
    v8i g4 = {0, 0, 0, 0, 0, 0, 0, 0}; // trailing group (6-arg toolchain variant)
    __builtin_amdgcn_tensor_load_to_lds(g0, g1, g2, g3, g4, 0);
}

// ---------------------------------------------------------------------------
// Kernel C: per-feature GRN experts with V_WMMA_F32_16X16X4_F32, TDM weight
// staging, and weighted accumulation into 'selected' via f32 atomics.
// grid = (NT/TOK_PER_WG, FD/FPG), block = 256 (8 waves, 1 token-tile each)
// ---------------------------------------------------------------------------
__global__ __launch_bounds__(WAVES * 32, 2)
void vsn_expert_kernel(const float* __restrict__ x,
                       const float* __restrict__ vw1,  const float* __restrict__ vb1,
                       const float* __restrict__ vw2,  const float* __restrict__ vb2,
                       const float* __restrict__ vwg,  const float* __restrict__ vbg,
                       const float* __restrict__ vwsk, const float* __restrict__ vbsk,
                       const float* __restrict__ vlng, const float* __restrict__ vlnb,
                       const float* __restrict__ wsel,     // [NT, FD] selection weights
                       float* __restrict__ outsel) {       // [NT, HD] accumulated
    __shared__ float sW2[HD * WSTRIDE];                    // 66 KB, TDM-padded
    __shared__ float sWg[HD * WSTRIDE];                    // 66 KB, TDM-padded
    __shared__ float sStage[WAVES][16 * WSTRIDE];          // 66 KB vh2 staging
    __shared__ float sVec[8][HD];                          // w1,b1,b2,bg,wsk,bsk,lng,lnb

    const int tid  = threadIdx.x;
    const int wave = tid >> 5;
    const int lane = tid & 31;
    const int col  = lane & 15;    // N index / M row index within half
    const int half = lane >> 4;    // K-half selector for A/B fragments
    const int t0   = blockIdx.x * TOK_PER_WG + wave * 16;
    const int f0   = blockIdx.y * FPG;
    float* const stg = &sStage[wave][0];

    v8f accsel[8];
#pragma unroll
    for (int n = 0; n < 8; ++n) accsel[n] = (v8f){0.f,0.f,0.f,0.f,0.f,0.f,0.f,0.f};

    for (int fi = 0; fi < FPG; ++fi) {
        const int f = f0 + fi;
        __syncthreads();           // previous iteration done reading weight LDS

        // --- stage this feature's weights: TDM for matrices, VALU for vectors ---
        if (tid < 32) {
            tdm_load_128x128(vw2 + (size_t)f * HD * HD, (unsigned)(uintptr_t)&sW2[0]);
            tdm_load_128x128(vwg + (size_t)f * HD * HD, (unsigned)(uintptr_t)&sWg[0]);
            __builtin_amdgcn_s_wait_tensorcnt(0);
        }
        if (tid < HD) {
            const int j = tid;     // 128 elements per small vector
            sVec[0][j] = vw1 [(size_t)f * HD + j];
            sVec[1][j] = vb1 [(size_t)f * HD + j];
            sVec[2][j] = vb2 [(size_t)f * HD + j];
            sVec[3][j] = vbg [(size_t)f * HD + j];
            sVec[4][j] = vwsk[(size_t)f * HD + j];
            sVec[5][j] = vbsk[(size_t)f * HD + j];
            sVec[6][j] = vlng[(size_t)f * HD + j];
            sVec[7][j] = vlnb[(size_t)f * HD + j];
        }
        __syncthreads();

        // x value for this lane's A-matrix row (rows M = lane&15, both halves)
        const float xv = x[(size_t)(t0 + col) * FD + f];

        v8f c[8];
#pragma unroll
        for (int n = 0; n < 8; ++n) c[n] = (v8f){0.f,0.f,0.f,0.f,0.f,0.f,0.f,0.f};

        // ---- GEMM1: vh2 = elu(x*w1+b1) @ W2^T   (16 x 128 = 8 col-tiles) ----
        // Unroll capped: 8 independent WMMA chains per step give enough ILP;
        // full unroll blows past 256 VGPRs (s_set_vgpr_msb churn, 1 wave/SIMD).
#pragma unroll 2
        for (int k = 0; k < HD; k += 4) {
            const int klo = k + half * 2;
            v2f a;
            a.x = elu1(xv * sVec[0][klo]     + sVec[1][klo]);
            a.y = elu1(xv * sVec[0][klo + 1] + sVec[1][klo + 1]);
#pragma unroll
            for (int n = 0; n < 8; ++n) {
                const int g = n * 16 + col;
                v2f b = *(const v2f*)&sW2[g * WSTRIDE + klo];
                c[n] = __builtin_amdgcn_wmma_f32_16x16x4_f32(
                    false, a, false, b, (short)0, c[n], false, false);
            }
        }

        // store vh2 (+bias b2) to padded LDS staging (C layout -> row major)
#pragma unroll
        for (int n = 0; n < 8; ++n) {
            const int g = n * 16 + col;
            const float b2v = sVec[2][g];
#pragma unroll
            for (int r = 0; r < 8; ++r)
                stg[(r + 8 * half) * WSTRIDE + g] = c[n][r] + b2v;
        }

        // ---- GEMM2: gate_pre = vh2 @ Wg^T ----
#pragma unroll
        for (int n = 0; n < 8; ++n) c[n] = (v8f){0.f,0.f,0.f,0.f,0.f,0.f,0.f,0.f};
#pragma unroll 2
        for (int k = 0; k < HD; k += 4) {
            const int klo = k + half * 2;
            v2f a = *(const v2f*)&stg[col * WSTRIDE + klo];
#pragma unroll
            for (int n = 0; n < 8; ++n) {
                const int g = n * 16 + col;
                v2f b = *(const v2f*)&sWg[g * WSTRIDE + klo];
                c[n] = __builtin_amdgcn_wmma_f32_16x16x4_f32(
                    false, a, false, b, (short)0, c[n], false, false);
            }
        }

        // ---- elementwise: sigmoid gate, skip, LayerNorm, weighted accumulate ----
#pragma unroll
        for (int r = 0; r < 8; ++r) {
            const int m = r + 8 * half;                 // token row in tile
            const float xm = x   [(size_t)(t0 + m) * FD + f];
            const float wm = wsel[(size_t)(t0 + m) * FD + f];
            float y[8];
            float s = 0.f, s2 = 0.f;
#pragma unroll
            for (int n = 0; n < 8; ++n) {
                const int g = n * 16 + col;
                const float vh2v = stg[m * WSTRIDE + g];
                const float vgv  = sigm(c[n][r] + sVec[3][g]);
                const float sk   = xm * sVec[4][g] + sVec[5][g];
                const float t    = vgv * vh2v + sk;
                y[n] = t; s += t; s2 += t * t;
            }
            // reduce over the 16 lanes of this half (128 values per token row)
#pragma unroll
            for (int msk = 1; msk < 16; msk <<= 1) {
                s  += __shfl_xor(s,  msk, 32);
                s2 += __shfl_xor(s2, msk, 32);
            }
            const float mean = s * (1.f / HD);
            const float var  = s2 * (1.f / HD) - mean * mean;
            const float rstd = __frsqrt_rn(var + LNEPS);
#pragma unroll
            for (int n = 0; n < 8; ++n) {
                const int g = n * 16 + col;
                const float vout = (y[n] - mean) * rstd * sVec[6][g] + sVec[7][g];
                accsel[n][r] += vout * wm;
            }
        }
    }

    // ---- reduce over feature groups into global output ----
#pragma unroll
    for (int n = 0; n < 8; ++n) {
#pragma unroll
        for (int r = 0; r < 8; ++r) {
            const int m = r + 8 * half;
            atomicAdd(&outsel[(size_t)(t0 + m) * HD + n * 16 + col], accsel[n][r]);
        }
    }
}

// ---------------------------------------------------------------------------
extern "C" void kernel_launch(void* const* d_in, const int* in_sizes, int n_in,
                              void* d_out, int out_size, void* d_ws, size_t ws_size,
                              hipStream_t stream) {
    (void)in_sizes; (void)n_in; (void)out_size; (void)d_ws; (void)ws_size;
    const float* x     = (const float*)d_in[0];
    const float* sw1   = (const float*)d_in[1];
    const float* sb1   = (const float*)d_in[2];
    const float* sw2   = (const float*)d_in[3];
    const float* sb2   = (const float*)d_in[4];
    const float* swg   = (const float*)d_in[5];
    const float* sbg   = (const float*)d_in[6];
    const float* slng  = (const float*)d_in[7];
    const float* slnb  = (const float*)d_in[8];
    const float* vw1   = (const float*)d_in[9];
    const float* vb1   = (const float*)d_in[10];
    const float* vw2   = (const float*)d_in[11];
    const float* vb2   = (const float*)d_in[12];
    const float* vwg   = (const float*)d_in[13];
    const float* vbg   = (const float*)d_in[14];
    const float* vwsk  = (const float*)d_in[15];
    const float* vbsk  = (const float*)d_in[16];
    const float* vlng  = (const float*)d_in[17];
    const float* vlnb  = (const float*)d_in[18];

    float* out_sel = (float*)d_out;                 // [NT, HD]
    float* out_w   = out_sel + (size_t)NT * HD;     // [NT, FD]

    vsn_zero_kernel<<<(NT * HD + 255) / 256, 256, 0, stream>>>(out_sel, NT * HD);

    vsn_sel_kernel<<<NT, 128, 0, stream>>>(x, sw1, sb1, sw2, sb2, swg, sbg,
                                           slng, slnb, out_w);

    dim3 grid2(NT / TOK_PER_WG, FD / FPG);
    vsn_expert_kernel<<<grid2, WAVES * 32, 0, stream>>>(
        x, vw1, vb1, vw2, vb2, vwg, vbg, vwsk, vbsk, vlng, vlnb, out_w, out_sel);
}